// GyroLoss_635655160326
// MI455X (gfx1250) — compile-verified
//
#include <hip/hip_runtime.h>
#include <hip/hip_bf16.h>
#include <stdint.h>

// ---------------- problem constants (from reference) ----------------
#define DT        0.01f
#define INV_HUBER 200.0f          // 1/0.005
// W*HUBER^2 = 1e6 * 2.5e-5 = 25
// level-4 mean denom: 64*2043*3 = 392256 ; level-5: 64*1019*3 = 195648 (x0.5 weight)
#define C4 (25.0f / 392256.0f)
#define C5 (25.0f / (2.0f * 195648.0f))

#define TPB        256
#define NBLOCKS    256            // 65536 level-5 outputs total
#define PAD_FLOATS 52             // 48 data floats + 4 pad (bank-conflict break)

// ---------------- SO(3) helpers ----------------

// General-angle exp (used for Xs, theta up to ~6)
__device__ __forceinline__ void so3_exp3(float x, float y, float z, float* R) {
  float t2 = x * x + y * y + z * z;
  float th = sqrtf(t2);
  float a, b;
  if (th < 1e-8f) {
    a = 1.0f; b = 0.5f;
  } else {
    float s, c;
    sincosf(th, &s, &c);
    a = s / th;
    b = (1.0f - c) / t2;
  }
  R[0] = 1.0f - b * (y * y + z * z);
  R[1] = -a * z + b * x * y;
  R[2] =  a * y + b * x * z;
  R[3] =  a * z + b * x * y;
  R[4] = 1.0f - b * (x * x + z * z);
  R[5] = -a * x + b * y * z;
  R[6] = -a * y + b * x * z;
  R[7] =  a * x + b * y * z;
  R[8] = 1.0f - b * (x * x + y * y);
}

// Small-angle exp for the Omega chain: theta = 0.01*||N(0,1)_3|| <= ~0.07.
// Taylor in t2 = theta^2: error ~ theta^8/9! ~ 1e-13, far below f32 ulp.
// Also reproduces the reference's theta<1e-8 guard (a->1, b->0.5) exactly.
__device__ __forceinline__ void so3_exp_small(float x, float y, float z, float* R) {
  float t2 = x * x + y * y + z * z;
  float a = 1.0f + t2 * (-1.0f / 6.0f  + t2 * ( 1.0f / 120.0f - t2 * (1.0f / 5040.0f)));
  float b = 0.5f + t2 * (-1.0f / 24.0f + t2 * ( 1.0f / 720.0f - t2 * (1.0f / 40320.0f)));
  R[0] = 1.0f - b * (y * y + z * z);
  R[1] = -a * z + b * x * y;
  R[2] =  a * y + b * x * z;
  R[3] =  a * z + b * x * y;
  R[4] = 1.0f - b * (x * x + z * z);
  R[5] = -a * x + b * y * z;
  R[6] = -a * y + b * x * z;
  R[7] =  a * x + b * y * z;
  R[8] = 1.0f - b * (x * x + y * y);
}

__device__ __forceinline__ void mm3(const float* A, const float* B, float* C) {
#pragma unroll
  for (int i = 0; i < 3; ++i) {
    float a0 = A[3 * i], a1 = A[3 * i + 1], a2 = A[3 * i + 2];
    C[3 * i + 0] = fmaf(a0, B[0], fmaf(a1, B[3], a2 * B[6]));
    C[3 * i + 1] = fmaf(a0, B[1], fmaf(a1, B[4], a2 * B[7]));
    C[3 * i + 2] = fmaf(a0, B[2], fmaf(a1, B[5], a2 * B[8]));
  }
}

// r = so3_log(O^T @ X)
__device__ __forceinline__ void log_bmtm(const float* O, const float* X, float* r) {
  float tr = 0.f, m21 = 0.f, m12 = 0.f, m02 = 0.f, m20 = 0.f, m10 = 0.f, m01 = 0.f;
#pragma unroll
  for (int k = 0; k < 3; ++k) {
    float o0 = O[3 * k], o1 = O[3 * k + 1], o2 = O[3 * k + 2];
    float x0 = X[3 * k], x1 = X[3 * k + 1], x2 = X[3 * k + 2];
    tr  = fmaf(o0, x0, fmaf(o1, x1, fmaf(o2, x2, tr)));
    m21 = fmaf(o2, x1, m21);  m12 = fmaf(o1, x2, m12);
    m02 = fmaf(o0, x2, m02);  m20 = fmaf(o2, x0, m20);
    m10 = fmaf(o1, x0, m10);  m01 = fmaf(o0, x1, m01);
  }
  float cs = 0.5f * (tr - 1.0f);
  cs = fminf(fmaxf(cs, -1.0f + 1e-7f), 1.0f - 1e-7f);
  float ang  = acosf(cs);
  float coef = (ang < 1e-6f) ? 0.5f : 0.5f * ang / sinf(ang);
  r[0] = coef * (m21 - m12);
  r[1] = coef * (m02 - m20);
  r[2] = coef * (m10 - m01);
}

__device__ __forceinline__ float huber3(const float* r) {
  float s = 0.f;
#pragma unroll
  for (int c = 0; c < 3; ++c) {
    float x = fabsf(r[c]) * INV_HUBER;
    s += (x < 1.0f) ? 0.5f * x * x : (x - 0.5f);
  }
  return s;
}

// chain-multiply 16 consecutive exp(DT*h) from LDS (sequential == tree product)
__device__ __forceinline__ void build_chain16(const float* hp, float* Om) {
  float M[9], T[9];
  so3_exp_small(DT * hp[0], DT * hp[1], DT * hp[2], Om);
#pragma unroll 2
  for (int k = 1; k < 16; ++k) {
    so3_exp_small(DT * hp[3 * k], DT * hp[3 * k + 1], DT * hp[3 * k + 2], M);
    mm3(Om, M, T);
#pragma unroll
    for (int j = 0; j < 9; ++j) Om[j] = T[j];
  }
}

// Stage one 16-sample-per-thread pass of hat data into an LDS buffer,
// coalesced 16B chunks via gfx1250 async global->LDS loads.
// pass p in {0,1}; thread u's data lands at ldsbase + u*PAD_FLOATS*4 .. +191.
__device__ __forceinline__ void stage_pass_async(const float* __restrict__ hat, int bid,
                                                 int tid, unsigned lbase, int p) {
#pragma unroll
  for (int i = 0; i < 12; ++i) {
    int c = tid + i * TPB;          // chunk id 0..3071
    int u = c / 12;                 // owning thread
    int q = c - u * 12;             // 16B piece within thread's 192B region
    size_t gb = (size_t)bid * 98304u + (size_t)u * 384u + (size_t)p * 192u + (size_t)q * 16u;
#if defined(__HIP_DEVICE_COMPILE__) && defined(__gfx1250__)
    const char* ga = (const char*)hat + gb;
    unsigned loff  = lbase + (unsigned)u * (PAD_FLOATS * 4u) + (unsigned)q * 16u;
    asm volatile("global_load_async_to_lds_b128 %0, %1, off"
                 :: "v"(loff), "v"(ga)
                 : "memory");
#else
    (void)lbase; (void)gb;
#endif
  }
}

// Wait until at most N async ops remain outstanding (N is an immediate).
template <int N>
__device__ __forceinline__ void wait_async() {
#if defined(__HIP_DEVICE_COMPILE__) && defined(__gfx1250__)
#if __has_builtin(__builtin_amdgcn_s_wait_asynccnt)
  __builtin_amdgcn_s_wait_asynccnt(N);
#else
  asm volatile("s_wait_asynccnt %0" :: "i"(N) : "memory");
#endif
#endif
}

__global__ __launch_bounds__(TPB)
void gyro_main(const float* __restrict__ xs, const float* __restrict__ hat,
               float* __restrict__ partial) {
  __shared__ float tile[2][TPB * PAD_FLOATS];  // 2 x 53248 B (double buffer)
  __shared__ float red[TPB];

  const int tid = threadIdx.x;
  const int bid = blockIdx.x;
  const int g   = bid * TPB + tid;          // level-5 output id, 0..65535
  // generic LDS addr low 32b == workgroup-relative LDS offset (aperture layout)
  const unsigned lbase0 = (unsigned)(uintptr_t)(&tile[0][0]);
  const unsigned lbase1 = (unsigned)(uintptr_t)(&tile[1][0]);

#if !(defined(__HIP_DEVICE_COMPILE__) && defined(__gfx1250__))
  // Portable fallback: synchronous staging
  auto stage_sync = [&](int p, int buf) {
#pragma unroll 1
    for (int i = 0; i < 12; ++i) {
      int c = tid + i * TPB;
      int u = c / 12;
      int q = c - u * 12;
      size_t gf = (size_t)bid * 24576u + (size_t)u * 96u + (size_t)p * 48u + (size_t)q * 4u;
      float4 v = *(const float4*)(hat + gf);
      *(float4*)&tile[buf][u * PAD_FLOATS + q * 4] = v;
    }
  };
#endif

  float OmA[9], OmB[9], XA[9], XB[9], r[3];

  // ---- issue BOTH staging passes asynchronously (24 in-flight async ops) ----
#if defined(__HIP_DEVICE_COMPILE__) && defined(__gfx1250__)
  stage_pass_async(hat, bid, tid, lbase0, 0);   // 12 ops -> buffer 0
  stage_pass_async(hat, bid, tid, lbase1, 1);   // 12 ops -> buffer 1
  wait_async<12>();        // async loads complete in order: buffer 0 is ready
#else
  stage_sync(0, 0);
  stage_sync(1, 1);
#endif
  __syncthreads();         // every wave's pass-0 writes are visible

  // ---- Xs loads overlap with in-flight buffer-1 traffic ----
  {
    size_t siA = (size_t)16 * (2 * (size_t)g);
    size_t siB = (size_t)16 * (2 * (size_t)g + 1);
    float4 qa = *(const float4*)(xs + 3 * siA);  // 192B-aligned, in-bounds
    float4 qb = *(const float4*)(xs + 3 * siB);
    so3_exp3(qa.x, qa.y, qa.z, XA);
    so3_exp3(qb.x, qb.y, qb.z, XB);
  }

  // ---- chain A from buffer 0 while buffer 1 loads land ----
  build_chain16(&tile[0][tid * PAD_FLOATS], OmA);

  wait_async<0>();         // buffer 1 ready (this wave)
  __syncthreads();         // ... and all other waves' writes too
  build_chain16(&tile[1][tid * PAD_FLOATS], OmB);

  // ---- level-4 residuals (rows of 2048, skip first 5 cols) ----
  float acc = 0.0f;
  {
    int jA = 2 * g, jB = 2 * g + 1;
    log_bmtm(OmA, XA, r);
    if ((jA & 2047) >= 5) acc += C4 * huber3(r);
    log_bmtm(OmB, XB, r);
    if ((jB & 2047) >= 5) acc += C4 * huber3(r);
  }

  // ---- level-5: pair-reduce once more (rows of 1024) ----
  {
    float Om5[9], X5[9];
    mm3(OmA, OmB, Om5);
    mm3(XA, XB, X5);
    log_bmtm(Om5, X5, r);
    if ((g & 1023) >= 5) acc += C5 * huber3(r);
  }

  // ---- deterministic block reduction ----
  red[tid] = acc;
  __syncthreads();
#pragma unroll
  for (int s = TPB / 2; s > 0; s >>= 1) {
    if (tid < s) red[tid] += red[tid + s];
    __syncthreads();
  }
  if (tid == 0) partial[bid] = red[0];
}

__global__ __launch_bounds__(TPB)
void gyro_reduce(const float* __restrict__ partial, float* __restrict__ out) {
  __shared__ float red[TPB];
  int tid = threadIdx.x;
  red[tid] = partial[tid];   // NBLOCKS == TPB
  __syncthreads();
#pragma unroll
  for (int s = TPB / 2; s > 0; s >>= 1) {
    if (tid < s) red[tid] += red[tid + s];
    __syncthreads();
  }
  if (tid == 0) out[0] = red[0];
}

extern "C" void kernel_launch(void* const* d_in, const int* in_sizes, int n_in,
                              void* d_out, int out_size, void* d_ws, size_t ws_size,
                              hipStream_t stream) {
  (void)in_sizes; (void)n_in; (void)out_size; (void)ws_size;
  const float* xs  = (const float*)d_in[0];
  const float* hat = (const float*)d_in[1];
  float* partial   = (float*)d_ws;     // NBLOCKS floats
  float* out       = (float*)d_out;

  gyro_main<<<dim3(NBLOCKS), dim3(TPB), 0, stream>>>(xs, hat, partial);
  gyro_reduce<<<dim3(1), dim3(TPB), 0, stream>>>(partial, out);
}